// MLFFNet_59141699666293
// MI455X (gfx1250) — compile-verified
//
#include <hip/hip_runtime.h>

// Problem constants (match reference)
#define B_      8
#define NATOM   512
#define NFEAT   42
#define NNEIGH  100
#define NPT     256
#define H1      64
#define H2      32
#define W0PAD   48      // NFEAT padded for both K (layer1) and N (dE layer)

typedef __attribute__((ext_vector_type(2))) float v2f;
typedef __attribute__((ext_vector_type(8))) float v8f;

// D(16x16,f32) = A(16x4,f32) x B(4x16,f32) + C  -- CDNA5 V_WMMA_F32_16X16X4_F32
static __device__ __forceinline__ v8f wmma4(v2f a, v2f b, v8f c) {
  return __builtin_amdgcn_wmma_f32_16x16x4_f32(
      /*neg_a=*/false, a, /*neg_b=*/false, b,
      /*c_mod=*/(short)0, c, /*reuse_a=*/false, /*reuse_b=*/false);
}

// gfx1250 hardware tanh (V_TANH_F32, single TRANS32 op) with safe fallback.
static __device__ __forceinline__ float fast_tanh(float x) {
#if __has_builtin(__builtin_amdgcn_tanhf)
  return __builtin_amdgcn_tanhf(x);
#else
  return tanhf(x);
#endif
}

// A-fragment (16x4 f32) from row-major LDS matrix M, leading dim ld.
// lane l: row m = l&15 ; holds K = kbase + 2*(l>=16) + {0,1}
static __device__ __forceinline__ v2f afrag(const float* M, int ld, int kbase, int l) {
  const int m  = l & 15;
  const int ko = (l >> 4) << 1;
  v2f a;
  a.x = M[m * ld + kbase + ko];
  a.y = M[m * ld + kbase + ko + 1];
  return a;
}

// ---------------------------------------------------------------------------
// Kernel 1: per-16-atom-tile MLP forward + analytic input-gradient, via WMMA.
// grid = B*NATOM/16 blocks, 32 threads (1 wave) each.
// All weights staged zero-padded in LDS -> unguarded ds_load_b64 B-fragments.
// ---------------------------------------------------------------------------
__global__ __launch_bounds__(32)
void mlp_fwd_bwd(const float* __restrict__ image,
                 const float* __restrict__ W0, const float* __restrict__ b0,
                 const float* __restrict__ W1, const float* __restrict__ b1,
                 const float* __restrict__ W2, const float* __restrict__ b2,
                 float* __restrict__ Ei_out, float* __restrict__ dE_out) {
  __shared__ float W0L[H1 * W0PAD];     // W0[j][f], f zero-padded 42->48
  __shared__ float W1L[H2 * H1];        // W1[j][k] verbatim
  __shared__ float XL [16 * 44];        // x, K padded 42->44
  __shared__ float H1L[16 * H1];        // tanh layer-1
  __shared__ float T1L[16 * H1];        // 1 - h1^2
  __shared__ float H2L[16 * H2];        // tanh layer-2
  __shared__ float T2L[16 * H2];        // 1 - h2^2
  __shared__ float G2L[16 * H2];        // g2 = W2 .* (1-h2^2)
  __shared__ float G1L[16 * H1];        // g1

  const int l    = threadIdx.x;
  const int gid0 = blockIdx.x * 16;     // first atom of tile (flat over B*NATOM)
  const int b    = gid0 / NATOM;
  const int a0   = gid0 % NATOM;
  const int t    = a0 / NPT;            // NPT multiple of 16 -> one type per tile

  const float* W0t = W0 + t * H1 * NFEAT;
  const float* b0t = b0 + t * H1;
  const float* W1t = W1 + t * H2 * H1;
  const float* b1t = b1 + t * H2;
  const float* W2t = W2 + t * H2;
  const float  b2s = b2[t];

  // stage weights (zero-padded) into LDS
  for (int i = l; i < H1 * W0PAD; i += 32) {
    const int j = i / W0PAD, f = i - j * W0PAD;
    W0L[i] = (f < NFEAT) ? W0t[j * NFEAT + f] : 0.f;
  }
  for (int i = l; i < H2 * H1; i += 32) W1L[i] = W1t[i];

  // load X tile (16 x 42), zero-pad K to 44
  const float* xin = image + ((size_t)b * NATOM + a0) * NFEAT;
  for (int i = l; i < 16 * 44; i += 32) {
    const int m = i / 44, f = i - m * 44;
    XL[i] = (f < NFEAT) ? xin[m * NFEAT + f] : 0.f;
  }
  __syncthreads();

  const int nl = l & 15;                // D-frag / B-frag column
  const int mh = (l >> 4) * 8;          // D-frag row base
  const int ko = (l >> 4) << 1;         // per-lane K sub-offset

  // ---- Layer 1: pre1(16x64) = X(16x42) @ W0^T(42x64) ----------------------
  for (int nt = 0; nt < 4; ++nt) {
    const int ncol = nt * 16 + nl;
    v8f acc = {};
    for (int kt = 0; kt < 11; ++kt) {   // K padded to 44 (zeros in both operands)
      const int k = kt * 4 + ko;
      v2f a = afrag(XL, 44, kt * 4, l);
      v2f bb;                           // B[k][n] = W0[n][k]
      bb.x = W0L[ncol * W0PAD + k];
      bb.y = W0L[ncol * W0PAD + k + 1];
      acc = wmma4(a, bb, acc);
    }
    const float bias = b0t[ncol];
    for (int r = 0; r < 8; ++r) {
      const float h = fast_tanh(acc[r] + bias);
      const int   m = mh + r;
      H1L[m * H1 + ncol] = h;
      T1L[m * H1 + ncol] = 1.f - h * h;
    }
  }
  __syncthreads();

  // ---- Layer 2: pre2(16x32) = H1(16x64) @ W1^T(64x32) ---------------------
  for (int nt = 0; nt < 2; ++nt) {
    const int ncol = nt * 16 + nl;
    v8f acc = {};
    for (int kt = 0; kt < 16; ++kt) {
      const int k = kt * 4 + ko;
      v2f a = afrag(H1L, H1, kt * 4, l);
      v2f bb;                           // B[k][n] = W1[n][k]
      bb.x = W1L[ncol * H1 + k];
      bb.y = W1L[ncol * H1 + k + 1];
      acc = wmma4(a, bb, acc);
    }
    const float bias = b1t[ncol];
    for (int r = 0; r < 8; ++r) {
      const float h = fast_tanh(acc[r] + bias);
      const int   m = mh + r;
      H2L[m * H2 + ncol] = h;
      T2L[m * H2 + ncol] = 1.f - h * h;
    }
  }
  __syncthreads();

  // ---- Layer 3 (Ei) + g2 --------------------------------------------------
  if (l < 16) {
    float ei = b2s;
    for (int j = 0; j < H2; ++j) ei += H2L[l * H2 + j] * W2t[j];
    Ei_out[(size_t)b * NATOM + a0 + l] = ei;
  }
  for (int i = l; i < 16 * H2; i += 32) {
    const int j = i & (H2 - 1);
    G2L[i] = T2L[i] * W2t[j];           // g2 = W2 .* (1 - h2^2)
  }
  __syncthreads();

  // ---- Backward: g1(16x64) = (1-h1^2) .* ( g2(16x32) @ W1(32x64) ) --------
  for (int nt = 0; nt < 4; ++nt) {
    const int ncol = nt * 16 + nl;
    v8f acc = {};
    for (int kt = 0; kt < 8; ++kt) {
      const int k = kt * 4 + ko;
      v2f a = afrag(G2L, H2, kt * 4, l);
      v2f bb;                           // B[k][n] = W1[k][n]
      bb.x = W1L[k       * H1 + ncol];
      bb.y = W1L[(k + 1) * H1 + ncol];
      acc = wmma4(a, bb, acc);
    }
    for (int r = 0; r < 8; ++r) {
      const int m = mh + r;
      G1L[m * H1 + ncol] = acc[r] * T1L[m * H1 + ncol];
    }
  }
  __syncthreads();

  // ---- Backward: dE(16x42) = g1(16x64) @ W0(64x42), N padded to 48 --------
  for (int nt = 0; nt < 3; ++nt) {
    const int ncol = nt * 16 + nl;
    v8f acc = {};
    for (int kt = 0; kt < 16; ++kt) {
      const int k = kt * 4 + ko;
      v2f a = afrag(G1L, H1, kt * 4, l);
      v2f bb;                           // B[k][f] = W0[k][f] (f>=42 reads zeros)
      bb.x = W0L[k       * W0PAD + ncol];
      bb.y = W0L[(k + 1) * W0PAD + ncol];
      acc = wmma4(a, bb, acc);
    }
    if (ncol < NFEAT) {
      for (int r = 0; r < 8; ++r) {
        const int m = mh + r;
        dE_out[((size_t)b * NATOM + a0 + m) * NFEAT + ncol] = acc[r];
      }
    }
  }
}

// ---------------------------------------------------------------------------
// Kernel 2: Etot[b] = sum_a Ei[b,a]   (deterministic tree reduce, no atomics)
// ---------------------------------------------------------------------------
__global__ __launch_bounds__(256)
void etot_kernel(const float* __restrict__ Ei, float* __restrict__ Etot) {
  __shared__ float s[256];
  const int b = blockIdx.x, tid = threadIdx.x;
  float v = 0.f;
  for (int i = tid; i < NATOM; i += 256) v += Ei[(size_t)b * NATOM + i];
  s[tid] = v; __syncthreads();
  for (int off = 128; off > 0; off >>= 1) {
    if (tid < off) s[tid] += s[tid + off];
    __syncthreads();
  }
  if (tid == 0) Etot[b] = s[0];
}

// ---------------------------------------------------------------------------
// Kernel 3: Force[b,a,k] = sum_{n,f} padded_dE[b, nbr[b,a,n], f] * dfeat[b,a,n,f,k]
// HBM-bound: streams 206 MB of dfeat once; dE (688 KB) gather hits L2.
// One block per atom (4096 blocks x 256 threads).
// ---------------------------------------------------------------------------
__global__ __launch_bounds__(256)
void force_kernel(const float* __restrict__ dE,
                  const float* __restrict__ dfeat,
                  const int*   __restrict__ neighbor,
                  float* __restrict__ Force) {
  __shared__ int   nbl[NNEIGH];
  __shared__ float red[3][256];
  const int tid = threadIdx.x;
  const int ba  = blockIdx.x;                 // b*NATOM + a
  const int b   = ba / NATOM;
  const int*   nb = neighbor + (size_t)ba * NNEIGH;
  const float* df = dfeat    + (size_t)ba * NNEIGH * NFEAT * 3;
  if (tid < NNEIGH) nbl[tid] = nb[tid];
  __syncthreads();

  float s0 = 0.f, s1 = 0.f, s2 = 0.f;
  for (int p = tid; p < NNEIGH * NFEAT; p += 256) {
    const int n = p / NFEAT;
    const int f = p - n * NFEAT;
    const int idx = nbl[n];                   // 0 => zero-padded row
    const float g = (idx == 0) ? 0.f
        : dE[((size_t)b * NATOM + (idx - 1)) * NFEAT + f];
    const float* dp = df + (size_t)p * 3;     // contiguous xyz
    // speculative prefetch one grid-stride ahead (lowers to global_prefetch_b8;
    // OOB speculative prefetches are silently dropped)
    __builtin_prefetch((const void*)(dp + 3 * 256), 0, 0);
    s0 = fmaf(g, dp[0], s0);
    s1 = fmaf(g, dp[1], s1);
    s2 = fmaf(g, dp[2], s2);
  }
  red[0][tid] = s0; red[1][tid] = s1; red[2][tid] = s2;
  __syncthreads();
  for (int off = 128; off > 0; off >>= 1) {
    if (tid < off) {
      red[0][tid] += red[0][tid + off];
      red[1][tid] += red[1][tid + off];
      red[2][tid] += red[2][tid + off];
    }
    __syncthreads();
  }
  if (tid == 0) {
    Force[(size_t)ba * 3 + 0] = red[0][0];
    Force[(size_t)ba * 3 + 1] = red[1][0];
    Force[(size_t)ba * 3 + 2] = red[2][0];
  }
}

// ---------------------------------------------------------------------------
extern "C" void kernel_launch(void* const* d_in, const int* in_sizes, int n_in,
                              void* d_out, int out_size, void* d_ws, size_t ws_size,
                              hipStream_t stream) {
  (void)in_sizes; (void)n_in; (void)out_size; (void)ws_size;
  const float* image    = (const float*)d_in[0];
  const float* dfeat    = (const float*)d_in[1];
  const int*   neighbor = (const int*)  d_in[2];   // harness delivers integers as int32
  // d_in[3] = natoms_img (unused by reference)
  const float* W0 = (const float*)d_in[4];
  const float* b0 = (const float*)d_in[5];
  const float* W1 = (const float*)d_in[6];
  const float* b1 = (const float*)d_in[7];
  const float* W2 = (const float*)d_in[8];
  const float* b2 = (const float*)d_in[9];

  float* out   = (float*)d_out;
  float* Etot  = out;                       // (B,1)
  float* Ei    = out + B_;                  // (B,NATOM,1)
  float* Force = out + B_ + B_ * NATOM;     // (B,NATOM,3)
  float* dE    = (float*)d_ws;              // (B,NATOM,NFEAT) scratch, 688 KB

  hipLaunchKernelGGL(mlp_fwd_bwd, dim3(B_ * NATOM / 16), dim3(32), 0, stream,
                     image, W0, b0, W1, b1, W2, b2, Ei, dE);
  hipLaunchKernelGGL(etot_kernel, dim3(B_), dim3(256), 0, stream, Ei, Etot);
  hipLaunchKernelGGL(force_kernel, dim3(B_ * NATOM), dim3(256), 0, stream,
                     dE, dfeat, neighbor, Force);
}